// GATConv_747324309915
// MI455X (gfx1250) — compile-verified
//
#include <hip/hip_runtime.h>
#include <math.h>

#define N_NODES_C 100000
#define N_EDGES_C 1600000
#define IN_SIZE   128
#define OUT_SIZE  16
#define NUM_HEADS 8
#define ZDIM      (OUT_SIZE * NUM_HEADS)   // 128
#define LEAKY     0.01f

// Exact-division properties relied on below (all true for this problem):
//   N_NODES_C % 16 == 0           -> no partial WMMA M tile
//   (N_NODES_C*NUM_HEADS) % 256 == 0
//   (N_NODES_C*ZDIM)      % 256 == 0
//   (N_EDGES_C*NUM_HEADS) % 256 == 0
//   N_EDGES_C % 8 == 0            -> whole waves per edge block
static_assert(N_NODES_C % 16 == 0, "full M tiles");
static_assert((N_NODES_C * NUM_HEADS) % 256 == 0, "exact grid");
static_assert((N_EDGES_C * NUM_HEADS) % 256 == 0, "exact grid");
static_assert(N_EDGES_C % 8 == 0, "exact grid");

typedef __attribute__((ext_vector_type(2))) float v2f;
typedef __attribute__((ext_vector_type(4))) float v4f;
typedef __attribute__((ext_vector_type(8))) float v8f;

// ---------------------------------------------------------------------------
// Kernel 1: Z = x @ W^T + b   via V_WMMA_F32_16X16X4_F32 (native fp32 WMMA)
// Block = 256 threads = 8 waves. One 16-node M tile per block; wave w owns
// channel tile [16w, 16w+16). K=128 swept in 32 steps of 4.
// fp32 A (16x4) layout: lanes 0-15 -> M=lane, K=k0+{0,1}; lanes 16-31 ->
// M=lane-16, K=k0+{2,3}. B (4x16) symmetric with N striped across lanes.
// All tiles are full (N_NODES_C % 16 == 0), so no guards: EXEC stays all-ones
// (WMMA requirement) and the store epilogue is 8 unconditional b32 stores.
// ---------------------------------------------------------------------------
__global__ void gat_gemm_wmma(const float* __restrict__ x,
                              const float* __restrict__ W,
                              const float* __restrict__ bias,
                              float* __restrict__ Z) {
    const int lane  = threadIdx.x & 31;
    const int wave  = threadIdx.x >> 5;          // 0..7  -> channel tile
    const int m0    = blockIdx.x * 16;           // node tile base
    const int c0    = wave * 16;                 // channel tile base
    const int mi    = lane & 15;
    const int khalf = (lane >> 4) * 2;           // 0 for lanes 0-15, 2 for 16-31

    const float* __restrict__ xrow = x + (size_t)(m0 + mi) * IN_SIZE;   // A: M=mi
    const float* __restrict__ wrow = W + (size_t)(c0 + mi) * IN_SIZE;   // B: N=mi

    v8f acc = {};
#pragma unroll 8
    for (int k0 = 0; k0 < IN_SIZE; k0 += 4) {
        const int k = k0 + khalf;
        v2f a, b;
        a.x = xrow[k];     a.y = xrow[k + 1];
        b.x = wrow[k];     b.y = wrow[k + 1];
        acc = __builtin_amdgcn_wmma_f32_16x16x4_f32(
            /*neg_a=*/false, a, /*neg_b=*/false, b,
            /*c_mod=*/(short)0, acc, /*reuse_a=*/false, /*reuse_b=*/false);
    }

    // C/D layout: VGPR v, lanes 0-15 -> M=v, lanes 16-31 -> M=v+8; N = lane&15.
    const float bv = bias[c0 + mi];
    float* __restrict__ zp =
        Z + (size_t)(m0 + ((lane >> 4) << 3)) * ZDIM + c0 + mi;
#pragma unroll
    for (int v = 0; v < 8; ++v)
        zp[(size_t)v * ZDIM] = acc[v] + bv;
}

// ---------------------------------------------------------------------------
// Kernel 2: per-node head scores  e_l[n,h] = sum_d Z[n,d*8+h]*a_l[d*8+h]
// ---------------------------------------------------------------------------
__global__ void gat_scores(const float* __restrict__ Z,
                           const float* __restrict__ a_l,
                           const float* __restrict__ a_r,
                           float* __restrict__ e_l,
                           float* __restrict__ e_r) {
    const int idx = blockIdx.x * blockDim.x + threadIdx.x;   // n*8 + h, exact grid
    const int n = idx >> 3, h = idx & 7;
    const float* __restrict__ zrow = Z + (size_t)n * ZDIM + h;
    float sl = 0.f, sr = 0.f;
#pragma unroll
    for (int d = 0; d < OUT_SIZE; ++d) {
        const float z = zrow[d * NUM_HEADS];
        sl += z * a_l[d * NUM_HEADS + h];
        sr += z * a_r[d * NUM_HEADS + h];
    }
    e_l[idx] = sl;
    e_r[idx] = sr;
}

// ---------------------------------------------------------------------------
// Kernel 3: zero out[], init m = -inf, s = 0 (d_out/d_ws are poisoned 0xAA)
// ---------------------------------------------------------------------------
__global__ void gat_init(float* __restrict__ out,
                         float* __restrict__ m,
                         float* __restrict__ s) {
    const int idx = blockIdx.x * blockDim.x + threadIdx.x;   // exact N*128 grid
    out[idx] = 0.0f;
    if (idx < N_NODES_C * NUM_HEADS) {
        m[idx] = -INFINITY;
        s[idx] = 0.0f;
    }
}

__device__ __forceinline__ float leaky(float a) {
    return a >= 0.0f ? a : LEAKY * a;
}

// ---------------------------------------------------------------------------
// Kernel 4: segment max  m[row,h] = max over edges  (GLOBAL_ATOMIC_MAX_NUM_F32)
// ---------------------------------------------------------------------------
__global__ void gat_edge_max(const int* __restrict__ row,
                             const int* __restrict__ col,
                             const float* __restrict__ e_l,
                             const float* __restrict__ e_r,
                             float* __restrict__ m) {
    const int idx = blockIdx.x * blockDim.x + threadIdx.x;   // e*8 + h, exact grid
    const int e = idx >> 3, h = idx & 7;
    const int r = row[e], c = col[e];
    const float a = leaky(e_l[r * NUM_HEADS + h] + e_r[c * NUM_HEADS + h]);
    atomicMax(&m[r * NUM_HEADS + h], a);
}

// ---------------------------------------------------------------------------
// Kernel 5: segment sum  s[row,h] += exp(a - m[row,h])
// ---------------------------------------------------------------------------
__global__ void gat_edge_sum(const int* __restrict__ row,
                             const int* __restrict__ col,
                             const float* __restrict__ e_l,
                             const float* __restrict__ e_r,
                             const float* __restrict__ m,
                             float* __restrict__ s) {
    const int idx = blockIdx.x * blockDim.x + threadIdx.x;   // exact grid
    const int e = idx >> 3, h = idx & 7;
    const int r = row[e], c = col[e];
    const float a = leaky(e_l[r * NUM_HEADS + h] + e_r[c * NUM_HEADS + h]);
    atomicAdd(&s[r * NUM_HEADS + h], __expf(a - m[r * NUM_HEADS + h]));
}

// ---------------------------------------------------------------------------
// Kernel 6: weighted scatter. One wave per edge; lane j owns 16B slice
// [4j,4j+4) of the 128-float row. The 4 elements of a slice share the head
// group h0 = (4j)&7 in {0,4}, so e_l/e_r/m/s are single float4 loads.
// Z gather is a contiguous 512B row (global_load_b128 per lane); out update
// is 4 f32 atomic adds per lane, absorbed by the 192MB L2 (out fits in L2).
// ---------------------------------------------------------------------------
__global__ void gat_scatter(const int* __restrict__ row,
                            const int* __restrict__ col,
                            const float* __restrict__ e_l,
                            const float* __restrict__ e_r,
                            const float* __restrict__ m,
                            const float* __restrict__ s,
                            const float* __restrict__ Z,
                            float* __restrict__ out) {
    const int lane = threadIdx.x & 31;
    const int e = blockIdx.x * (blockDim.x >> 5) + (threadIdx.x >> 5); // exact grid

    const int r  = row[e];
    const int c  = col[e];
    const int j4 = lane * 4;        // element base within row, 0..124
    const int h0 = j4 & 7;          // 0 or 4

    const v4f vl = *(const v4f*)(e_l + (size_t)r * NUM_HEADS + h0);
    const v4f vr = *(const v4f*)(e_r + (size_t)c * NUM_HEADS + h0);
    const v4f vm = *(const v4f*)(m   + (size_t)r * NUM_HEADS + h0);
    const v4f vs = *(const v4f*)(s   + (size_t)r * NUM_HEADS + h0);
    const v4f zz = *(const v4f*)(Z   + (size_t)c * ZDIM + j4);
    float* op = out + (size_t)r * ZDIM + j4;

#pragma unroll
    for (int t = 0; t < 4; ++t) {
        const float a   = leaky(vl[t] + vr[t]);
        const float att = __expf(a - vm[t]) / vs[t];
        atomicAdd(op + t, att * zz[t]);
    }
}

// ---------------------------------------------------------------------------
extern "C" void kernel_launch(void* const* d_in, const int* in_sizes, int n_in,
                              void* d_out, int out_size, void* d_ws, size_t ws_size,
                              hipStream_t stream) {
    const float* x   = (const float*)d_in[0];
    const float* W   = (const float*)d_in[1];
    const float* b   = (const float*)d_in[2];
    const float* a_l = (const float*)d_in[3];
    const float* a_r = (const float*)d_in[4];
    const int*   row = (const int*)d_in[5];
    const int*   col = (const int*)d_in[6];
    float* out = (float*)d_out;

    // Workspace layout (floats): Z [N*128] | e_l [N*8] | e_r [N*8] | m [N*8] | s [N*8]
    float* Z   = (float*)d_ws;
    float* e_l = Z   + (size_t)N_NODES_C * ZDIM;
    float* e_r = e_l + (size_t)N_NODES_C * NUM_HEADS;
    float* m   = e_r + (size_t)N_NODES_C * NUM_HEADS;
    float* s   = m   + (size_t)N_NODES_C * NUM_HEADS;

    (void)in_sizes; (void)n_in; (void)out_size; (void)ws_size;

    // 1) GEMM (WMMA fp32): 16 nodes x 128 channels per block
    gat_gemm_wmma<<<N_NODES_C / 16, 256, 0, stream>>>(x, W, b, Z);

    // 2) per-node head scores
    gat_scores<<<(N_NODES_C * NUM_HEADS) / 256, 256, 0, stream>>>(Z, a_l, a_r, e_l, e_r);

    // 3) init out / m / s
    gat_init<<<(N_NODES_C * ZDIM) / 256, 256, 0, stream>>>(out, m, s);

    // 4) segment max
    gat_edge_max<<<(N_EDGES_C * NUM_HEADS) / 256, 256, 0, stream>>>(row, col, e_l, e_r, m);

    // 5) segment sum
    gat_edge_sum<<<(N_EDGES_C * NUM_HEADS) / 256, 256, 0, stream>>>(row, col, e_l, e_r, m, s);

    // 6) weighted scatter: one wave per edge, 8 waves per block
    gat_scatter<<<N_EDGES_C / 8, 256, 0, stream>>>(row, col, e_l, e_r, m, s, Z, out);
}